// FB15K_XGradNet_32908039422281
// MI455X (gfx1250) — compile-verified
//
#include <hip/hip_runtime.h>
#include <math.h>

#define N_NODES 14541
#define NP      14560      /* padded to multiple of 32 for 32-row WMMA tiles */
#define R_REL   64
#define E_EDGES 16384
#define RC      4          /* relations per chunk */
#define NCHUNK  (R_REL / RC)

typedef __attribute__((ext_vector_type(2))) float v2f;
typedef __attribute__((ext_vector_type(8))) float v8f;

// ---------------- utility kernels ----------------

__global__ void zero_kernel(float* __restrict__ p, long n) {
  long i = (long)blockIdx.x * blockDim.x + threadIdx.x;
  if (i < n) p[i] = 0.0f;
}

__global__ void degree_kernel(const int* __restrict__ src, const int* __restrict__ dst,
                              float* __restrict__ deg_out, float* __restrict__ deg_in) {
  long i = (long)blockIdx.x * blockDim.x + threadIdx.x;
  if (i >= (long)R_REL * E_EDGES) return;
  long r = i / E_EDGES;
  atomicAdd(&deg_out[r * NP + src[i]], 1.0f);
  atomicAdd(&deg_in [r * NP + dst[i]], 1.0f);
}

__global__ void norm_kernel(float* __restrict__ p, long n) {
  long i = (long)blockIdx.x * blockDim.x + threadIdx.x;
  if (i < n) p[i] = rsqrtf(fmaxf(p[i], 1.0f));   // where(deg>0,deg,1)^-0.5
}

__global__ void bias_mean_kernel(const float* __restrict__ b, float* __restrict__ bm, int fout) {
  int f = blockIdx.x * blockDim.x + threadIdx.x;
  if (f >= fout) return;
  float s = 0.0f;
  for (int r = 0; r < R_REL; ++r) s += b[r * fout + f];
  bm[f] = s * (1.0f / R_REL);
}

__global__ void bias_init_kernel(const float* __restrict__ bm, float* __restrict__ out,
                                 int fout, long total) {
  long i = (long)blockIdx.x * blockDim.x + threadIdx.x;
  if (i < total) out[i] = bm[i % fout];
}

// WT[r][n][k] = W[r][k][n]  (so all WMMA B-loads become contiguous b64)
__global__ void transpose_w_kernel(const float* __restrict__ W, float* __restrict__ WT,
                                   int fout, int K, long total) {
  long i = (long)blockIdx.x * blockDim.x + threadIdx.x;
  if (i >= total) return;
  int  n  = (int)(i % fout);
  long rk = i / fout;
  int  k  = (int)(rk % K);
  long r  = rk / K;
  WT[(r * fout + n) * (long)K + k] = W[i];
}

// ---------------- graph scatter (gather-normalize-atomic-scatter) ----------------
// Both norms folded into the per-edge weight (linearity of segment_sum):
//   agg[d] = norm_in[d] * sum_e x[src_e]*norm_out[src_e]
// messages are 128 floats; 32 lanes per edge, float4 each => coalesced rows.

__global__ void scatter_kernel(const float* __restrict__ x, const int* __restrict__ src,
                               const int* __restrict__ dst, const float* __restrict__ norm_out,
                               const float* __restrict__ norm_in,
                               float* __restrict__ agg, int r_base, int do_relu) {
  long i = (long)blockIdx.x * blockDim.x + threadIdx.x;
  if (i >= (long)RC * E_EDGES * 32) return;
  int  f4 = (int)(i & 31);
  long ef = i >> 5;
  int  rl = (int)(ef / E_EDGES);
  int  e  = (int)(ef % E_EDGES);
  int  r  = r_base + rl;
  long eo = (long)r * E_EDGES + e;
  int  s  = src[eo];
  int  d  = dst[eo];
  float w = norm_out[(long)r * NP + s] * norm_in[(long)r * NP + d];
  const float4* xp = (const float4*)(x + (long)s * 128) + f4;
  float4 v = *xp;
  if (do_relu) { v.x = fmaxf(v.x, 0.f); v.y = fmaxf(v.y, 0.f);
                 v.z = fmaxf(v.z, 0.f); v.w = fmaxf(v.w, 0.f); }
  float* ap = agg + ((long)rl * NP + d) * 128 + f4 * 4;
  atomicAdd(ap + 0, v.x * w);
  atomicAdd(ap + 1, v.y * w);
  atomicAdd(ap + 2, v.z * w);
  atomicAdd(ap + 3, v.w * w);
}

// ---------------- fp32 WMMA GEMM ----------------
// C[m,n] += alpha * sum_r A_r[m,k] * BT_r[n,k]
// One wave32 owns a 32(M) x 32(N) tile: 4 accumulators, so each of the 4 b64
// loads per K-step feeds two V_WMMA_F32_16X16X4_F32 (4 wmma : 4 loads).
// Layouts (ISA 7.12.2):
//   A 16x4: M = lane&15, K = {k0,k0+1} (lanes 0-15) / {k0+2,k0+3} (lanes 16-31)
//   C/D:    row = m0 + v + 8*(lane>=16), col = n_tile + (lane&15)
// OOB tail rows: D-row m depends only on A-row m, so clamped (safe) pointers
// plus the row<rowsA store guard are sufficient; EXEC stays all-1s throughout.

template <int K>
__global__ void __launch_bounds__(128)
wmma_gemm_kernel(const float* __restrict__ A0, long strideA, int lda, int rowsA,
                 const float* __restrict__ BT0, long strideB, int ldbt,
                 float* __restrict__ C, int ldc,
                 int tiles_m, int tiles_n, int nr, float alpha) {
  int wave = (int)blockIdx.x * ((int)blockDim.x >> 5) + ((int)threadIdx.x >> 5);
  int lane = (int)threadIdx.x & 31;
  int tn = wave % tiles_n;
  int tm = wave / tiles_n;
  if (tm >= tiles_m) return;                 // wave-uniform exit
  int m0 = tm << 5, n0 = tn << 5;
  int mrow0 = m0 + (lane & 15);
  int mrow1 = mrow0 + 16;
  int ncol  = n0 + (lane & 15);
  int khalf = (lane >> 4) << 1;              // 0 or 2
  int mr0 = (mrow0 < rowsA) ? mrow0 : 0;     // clamp => loads always in-bounds
  int mr1 = (mrow1 < rowsA) ? mrow1 : 0;

  v8f acc00 = {}, acc01 = {}, acc10 = {}, acc11 = {};
#pragma unroll 1
  for (int r = 0; r < nr; ++r) {
    const float* a0 = A0 + (long)r * strideA + (long)mr0 * lda + khalf;
    const float* a1 = A0 + (long)r * strideA + (long)mr1 * lda + khalf;
    const float* b0 = BT0 + (long)r * strideB + (long)ncol * ldbt + khalf;
    const float* b1 = b0 + 16 * (long)ldbt;
#pragma unroll
    for (int k0 = 0; k0 < K; k0 += 4) {
      v2f va0 = *(const v2f*)(a0 + k0);
      v2f va1 = *(const v2f*)(a1 + k0);
      v2f vb0 = *(const v2f*)(b0 + k0);
      v2f vb1 = *(const v2f*)(b1 + k0);
      acc00 = __builtin_amdgcn_wmma_f32_16x16x4_f32(false, va0, false, vb0,
                                                    (short)0, acc00, false, false);
      acc01 = __builtin_amdgcn_wmma_f32_16x16x4_f32(false, va0, false, vb1,
                                                    (short)0, acc01, false, false);
      acc10 = __builtin_amdgcn_wmma_f32_16x16x4_f32(false, va1, false, vb0,
                                                    (short)0, acc10, false, false);
      acc11 = __builtin_amdgcn_wmma_f32_16x16x4_f32(false, va1, false, vb1,
                                                    (short)0, acc11, false, false);
    }
  }
  int rb0 = m0 + ((lane >> 4) << 3);
#pragma unroll
  for (int v = 0; v < 8; ++v) {
    int r0 = rb0 + v, r1 = r0 + 16;
    if (r0 < rowsA) {
      C[(long)r0 * ldc + ncol]      += alpha * acc00[v];
      C[(long)r0 * ldc + ncol + 16] += alpha * acc01[v];
    }
    if (r1 < rowsA) {
      C[(long)r1 * ldc + ncol]      += alpha * acc10[v];
      C[(long)r1 * ldc + ncol + 16] += alpha * acc11[v];
    }
  }
}

// ---------------- LSTM epilogue ----------------
// gates[n, d*128 + {i:0..31, f:32..63, g:64..95, o:96..127}]; f-gate dead (c0=0).

__global__ void gates_init_kernel(const float* __restrict__ b_ih, const float* __restrict__ b_hh,
                                  float* __restrict__ gates, long total) {
  long i = (long)blockIdx.x * blockDim.x + threadIdx.x;
  if (i >= total) return;
  int c = (int)(i & 255);                    // c = d*128 + j
  gates[i] = b_ih[c] + b_hh[c];
}

__global__ void lstm_kernel(const float* __restrict__ gates, float* __restrict__ out) {
  long i = (long)blockIdx.x * blockDim.x + threadIdx.x;
  if (i >= (long)N_NODES * 64) return;
  int n = (int)(i >> 6);
  int c = (int)(i & 63);                     // c = d*32 + u
  int d = c >> 5;
  int u = c & 31;
  const float* g = gates + (long)n * 256 + d * 128;
  float gi = g[u], gg = g[64 + u], go = g[96 + u];
  float cc = (1.0f / (1.0f + expf(-gi))) * tanhf(gg);
  out[i]   = (1.0f / (1.0f + expf(-go))) * tanhf(cc);
}

// ---------------- launch ----------------

static inline int cdiv(long a, int b) { return (int)((a + b - 1) / b); }

extern "C" void kernel_launch(void* const* d_in, const int* in_sizes, int n_in,
                              void* d_out, int out_size, void* d_ws, size_t ws_size,
                              hipStream_t stream) {
  (void)in_sizes; (void)n_in; (void)out_size; (void)ws_size;
  const int*   src  = (const int*)  d_in[1];
  const int*   dst  = (const int*)  d_in[2];
  const float* emb  = (const float*)d_in[3];
  const float* W1   = (const float*)d_in[4];
  const float* b1   = (const float*)d_in[5];
  const float* W2   = (const float*)d_in[6];
  const float* b2   = (const float*)d_in[7];
  const float* w_ih = (const float*)d_in[8];   // [2,128,64] == [d][n][k]; w_hh (d_in[9]) dead
  const float* b_ih = (const float*)d_in[10];
  const float* b_hh = (const float*)d_in[11];
  float* out = (float*)d_out;

  // workspace layout (floats), ~51 MB total
  float* ws       = (float*)d_ws;
  float* norm_out = ws;                                  // R*NP
  float* norm_in  = norm_out + (long)R_REL * NP;         // R*NP
  float* h1       = norm_in  + (long)R_REL * NP;         // NP*128
  float* agg      = h1       + (long)NP * 128;           // RC*NP*128 (reused as gates NP*256)
  float* bmean    = agg      + (long)RC * NP * 128;      // 128
  float* w1t      = bmean    + 128;                      // R*128*128
  float* w2t      = w1t      + (long)R_REL * 128 * 128;  // R*64*128
  float* gates    = agg;

  const int T = 256;
  const int TM = NP / 32;                                // 455 m-tiles (32 rows each)

  // 1) degrees -> norms (shared by both conv layers); transpose weights once
  long nnorm = 2L * R_REL * NP;
  zero_kernel<<<cdiv(nnorm, T), T, 0, stream>>>(norm_out, nnorm);
  degree_kernel<<<cdiv((long)R_REL * E_EDGES, T), T, 0, stream>>>(src, dst, norm_out, norm_in);
  norm_kernel<<<cdiv(nnorm, T), T, 0, stream>>>(norm_out, nnorm);
  transpose_w_kernel<<<cdiv((long)R_REL * 128 * 128, T), T, 0, stream>>>(
      W1, w1t, 128, 128, (long)R_REL * 128 * 128);
  transpose_w_kernel<<<cdiv((long)R_REL * 128 * 64, T), T, 0, stream>>>(
      W2, w2t, 64, 128, (long)R_REL * 128 * 64);

  // 2) conv1: h1 = mean_r(S_r x W1_r) + mean_r b1_r   (ReLU deferred to conv2 gather)
  bias_mean_kernel<<<1, 128, 0, stream>>>(b1, bmean, 128);
  bias_init_kernel<<<cdiv((long)NP * 128, T), T, 0, stream>>>(bmean, h1, 128, (long)NP * 128);
  for (int c = 0; c < NCHUNK; ++c) {
    int rb = c * RC;
    zero_kernel<<<cdiv((long)RC * NP * 128, T), T, 0, stream>>>(agg, (long)RC * NP * 128);
    scatter_kernel<<<cdiv((long)RC * E_EDGES * 32, T), T, 0, stream>>>(
        emb, src, dst, norm_out, norm_in, agg, rb, /*relu=*/0);
    wmma_gemm_kernel<128><<<cdiv((long)TM * 4, 4), 128, 0, stream>>>(
        agg, (long)NP * 128, 128, N_NODES,
        w1t + (long)rb * 128 * 128, (long)128 * 128, 128,
        h1, 128, TM, /*tiles_n=*/4, RC, 1.0f / R_REL);
  }

  // 3) conv2: h2 = mean_r(S_r relu(h1) W2_r) + mean_r b2_r -> stored in d_out
  bias_mean_kernel<<<1, 64, 0, stream>>>(b2, bmean, 64);
  bias_init_kernel<<<cdiv((long)N_NODES * 64, T), T, 0, stream>>>(bmean, out, 64, (long)N_NODES * 64);
  for (int c = 0; c < NCHUNK; ++c) {
    int rb = c * RC;
    zero_kernel<<<cdiv((long)RC * NP * 128, T), T, 0, stream>>>(agg, (long)RC * NP * 128);
    scatter_kernel<<<cdiv((long)RC * E_EDGES * 32, T), T, 0, stream>>>(
        h1, src, dst, norm_out, norm_in, agg, rb, /*relu=*/1);
    wmma_gemm_kernel<128><<<cdiv((long)TM * 2, 4), 128, 0, stream>>>(
        agg, (long)NP * 128, 128, N_NODES,
        w2t + (long)rb * 64 * 128, (long)64 * 128, 128,
        out, 64, TM, /*tiles_n=*/2, RC, 1.0f / R_REL);
  }

  // 4) BiLSTM (seq_len=1): gates = h2 @ w_ih[d]^T + b_ih + b_hh  (w_ih already [n][k])
  gates_init_kernel<<<cdiv((long)NP * 256, T), T, 0, stream>>>(b_ih, b_hh, gates, (long)NP * 256);
  for (int d = 0; d < 2; ++d) {
    wmma_gemm_kernel<64><<<cdiv((long)TM * 4, 4), 128, 0, stream>>>(
        out, 0, 64, N_NODES,
        w_ih + (long)d * 128 * 64, 0, 64,
        gates + d * 128, 256, TM, /*tiles_n=*/4, /*nr=*/1, 1.0f);
  }
  lstm_kernel<<<cdiv((long)N_NODES * 64, T), T, 0, stream>>>(gates, out);
}